// GCN_79740362817954
// MI455X (gfx1250) — compile-verified
//
#include <hip/hip_runtime.h>
#include <hip/hip_bf16.h>

typedef __attribute__((ext_vector_type(2))) float v2f;
typedef __attribute__((ext_vector_type(8))) float v8f;

// ---------------------------------------------------------------------------
// Degree / normalization kernels
// ---------------------------------------------------------------------------
__global__ void gcn_deg_init(float* __restrict__ deg, int n) {
    int i = blockIdx.x * blockDim.x + threadIdx.x;
    if (i < n) deg[i] = 1.0f;   // self-loop contributes 1 to every node
}

__global__ void gcn_deg_count(const int* __restrict__ dst, float* __restrict__ deg, int e) {
    int i = blockIdx.x * blockDim.x + threadIdx.x;
    if (i < e) unsafeAtomicAdd(&deg[dst[i]], 1.0f);
}

__global__ void gcn_dinv(const float* __restrict__ deg, float* __restrict__ dinv, int n) {
    int i = blockIdx.x * blockDim.x + threadIdx.x;
    if (i < n) {
        float d = deg[i];
        dinv[i] = (d > 0.0f) ? rsqrtf(d) : 0.0f;
    }
}

// ---------------------------------------------------------------------------
// Fused dense GEMM:
//   pre-op (optional): a = relu(a + bias[k])           (applied to A reads)
//   OutLin[r,c] = sum_k a[r,k] * W[k,c]
//   OutAgg[r,c] = OutLin[r,c] * dinv[r]^2              (self-loop seed)
// One wave computes a 16-row x 128-col strip via V_WMMA_F32_16X16X4_F32.
// Block = 256 threads = 8 waves = 128 rows. W pre-packed in LDS into the
// B-fragment layout so each (b.x,b.y) K-pair is one aligned ds_load_b64:
//   sWp[kk*512 + ct*64 + half*32 + l16*2 + j] = W[4*kk + 2*half + j][ct*16+l16]
// (lanes 0-15 -> banks 0-31, lanes 16-31 -> banks 32-63: conflict-free)
// ---------------------------------------------------------------------------
template <bool RELU>
__launch_bounds__(256)
__global__ void gcn_gemm128(const float* __restrict__ A,
                            const float* __restrict__ bias,
                            const float* __restrict__ W,
                            float* __restrict__ OutLin,
                            float* __restrict__ OutAgg,
                            const float* __restrict__ dinv,
                            int nrows) {
    __shared__ float sWp[128 * 128];         // 64 KB, fragment-packed
    const int tid = threadIdx.x;

    // cooperative load + pack of W — 16 float4 / thread, scalar scatter to LDS
    #pragma unroll
    for (int i = 0; i < 16; ++i) {
        int g = (i * 256 + tid) * 4;
        int k = g >> 7;
        int n = g & 127;
        float4 w4 = *(const float4*)(W + g);
        int kk = k >> 2, r = k & 3;
        int hf = r >> 1, j = r & 1;
        float wv[4] = {w4.x, w4.y, w4.z, w4.w};
        #pragma unroll
        for (int q = 0; q < 4; ++q) {
            int col = n + q;
            sWp[kk * 512 + (col >> 4) * 64 + hf * 32 + (col & 15) * 2 + j] = wv[q];
        }
    }
    __syncthreads();

    const int wave = tid >> 5;
    const int lane = tid & 31;
    const int half = lane >> 4;              // 0: lanes 0-15, 1: lanes 16-31
    const int l16  = lane & 15;
    const int rowtile = blockIdx.x * 8 + wave;
    if (rowtile * 16 >= nrows) return;       // wave-uniform (nrows % 16 == 0)

    v8f acc[8];
    #pragma unroll
    for (int ct = 0; ct < 8; ++ct)
        acc[ct] = (v8f){0.f, 0.f, 0.f, 0.f, 0.f, 0.f, 0.f, 0.f};

    const float* arow = A + (size_t)(rowtile * 16 + l16) * 128;

    for (int kk = 0; kk < 32; ++kk) {
        const int kb = kk * 4 + 2 * half;    // contiguous K pair per half-wave
        float2 a2 = *(const float2*)(arow + kb);
        if (RELU) {
            const float2 bb = *(const float2*)(bias + kb);
            a2.x = fmaxf(a2.x + bb.x, 0.0f);
            a2.y = fmaxf(a2.y + bb.y, 0.0f);
        }
        v2f a; a.x = a2.x; a.y = a2.y;
        const float* bbase = &sWp[kk * 512 + half * 32 + l16 * 2];
        #pragma unroll
        for (int ct = 0; ct < 8; ++ct) {
            const float2 bp = *(const float2*)(bbase + ct * 64);
            v2f b; b.x = bp.x; b.y = bp.y;
            acc[ct] = __builtin_amdgcn_wmma_f32_16x16x4_f32(
                false, a, false, b, (short)0, acc[ct], false, false);
        }
    }

    // D layout: VGPR j -> row 16*tile + j + 8*half, col = ct*16 + l16
    const int rowbase = rowtile * 16 + 8 * half;
    float d2[8];
    #pragma unroll
    for (int j = 0; j < 8; ++j) {
        float s = dinv[rowbase + j];
        d2[j] = s * s;
    }
    #pragma unroll
    for (int ct = 0; ct < 8; ++ct) {
        const int col = ct * 16 + l16;
        #pragma unroll
        for (int j = 0; j < 8; ++j) {
            const size_t off = (size_t)(rowbase + j) * 128 + col;
            const float v = acc[ct][j];
            OutLin[off] = v;
            OutAgg[off] = v * d2[j];
        }
    }
}

// ---------------------------------------------------------------------------
// Edge scatter: out[dst,:] += h[src,:] * dinv[src]*dinv[dst]  (one wave/edge)
// ---------------------------------------------------------------------------
__global__ void gcn_scatter(const int* __restrict__ src, const int* __restrict__ dst,
                            const float* __restrict__ dinv, const float* __restrict__ h,
                            float* __restrict__ out, int e) {
    int w = (blockIdx.x * blockDim.x + threadIdx.x) >> 5;
    int lane = threadIdx.x & 31;
    if (w >= e) return;
    int s = src[w];
    int d = dst[w];
    float nm = dinv[s] * dinv[d];
    float4 v = *(const float4*)(h + (size_t)s * 128 + lane * 4);
    float* o = out + (size_t)d * 128 + lane * 4;
    unsafeAtomicAdd(o + 0, v.x * nm);
    unsafeAtomicAdd(o + 1, v.y * nm);
    unsafeAtomicAdd(o + 2, v.z * nm);
    unsafeAtomicAdd(o + 3, v.w * nm);
}

// ---------------------------------------------------------------------------
// Output projection (128 -> 5) + softmax, with fused relu(h + b2) pre-op.
// One wave32 per node.
// ---------------------------------------------------------------------------
__global__ void gcn_out_softmax(const float* __restrict__ h, const float* __restrict__ b2,
                                const float* __restrict__ Wo, const float* __restrict__ bo,
                                float* __restrict__ out, int n) {
    int w = (blockIdx.x * blockDim.x + threadIdx.x) >> 5;
    int lane = threadIdx.x & 31;
    if (w >= n) return;
    float4 v = *(const float4*)(h + (size_t)w * 128 + lane * 4);
    const float4 bb = *(const float4*)(b2 + lane * 4);
    v.x = fmaxf(v.x + bb.x, 0.0f);
    v.y = fmaxf(v.y + bb.y, 0.0f);
    v.z = fmaxf(v.z + bb.z, 0.0f);
    v.w = fmaxf(v.w + bb.w, 0.0f);
    float acc[5];
    #pragma unroll
    for (int c = 0; c < 5; ++c) {
        acc[c] = v.x * Wo[(lane * 4 + 0) * 5 + c]
               + v.y * Wo[(lane * 4 + 1) * 5 + c]
               + v.z * Wo[(lane * 4 + 2) * 5 + c]
               + v.w * Wo[(lane * 4 + 3) * 5 + c];
    }
    #pragma unroll
    for (int c = 0; c < 5; ++c) {
        #pragma unroll
        for (int off = 16; off > 0; off >>= 1)
            acc[c] += __shfl_xor(acc[c], off, 32);
    }
    // every lane now holds all 5 reduced logits
    #pragma unroll
    for (int c = 0; c < 5; ++c) acc[c] += bo[c];
    float m = acc[0];
    #pragma unroll
    for (int c = 1; c < 5; ++c) m = fmaxf(m, acc[c]);
    float ex[5], s = 0.0f;
    #pragma unroll
    for (int c = 0; c < 5; ++c) { ex[c] = expf(acc[c] - m); s += ex[c]; }
    float inv = 1.0f / s;
    if (lane == 0) {
        #pragma unroll
        for (int c = 0; c < 5; ++c) out[(size_t)w * 5 + c] = ex[c] * inv;
    }
}

// ---------------------------------------------------------------------------
// Launch
// ---------------------------------------------------------------------------
extern "C" void kernel_launch(void* const* d_in, const int* in_sizes, int n_in,
                              void* d_out, int out_size, void* d_ws, size_t ws_size,
                              hipStream_t stream) {
    const float* x  = (const float*)d_in[0];
    const int*   ei = (const int*)d_in[1];
    const float* W1 = (const float*)d_in[2];
    const float* b1 = (const float*)d_in[3];
    const float* W2 = (const float*)d_in[4];
    const float* b2 = (const float*)d_in[5];
    const float* Wo = (const float*)d_in[6];
    const float* bo = (const float*)d_in[7];

    const int n = in_sizes[0] / 128;     // 50000
    const int e = in_sizes[1] / 2;       // 600000
    const int* srcp = ei;
    const int* dstp = ei + e;

    float* deg  = (float*)d_ws;
    float* dinv = deg + n;
    float* bufA = dinv + n;                       // linear-transform result
    float* bufB = bufA + (size_t)n * 128;         // aggregated result
    float* outp = (float*)d_out;

    const int T = 256;
    const int blk_n    = (n + T - 1) / T;
    const int blk_e    = (e + T - 1) / T;
    const int blk_ew   = (int)(((size_t)e * 32 + T - 1) / T);   // wave-per-edge
    const int blk_gemm = (((n + 15) / 16) + 7) / 8;             // 8 row-tiles / block
    const int blk_ow   = (int)(((size_t)n * 32 + T - 1) / T);   // wave-per-node

    // normalization
    gcn_deg_init<<<blk_n, T, 0, stream>>>(deg, n);
    gcn_deg_count<<<blk_e, T, 0, stream>>>(dstp, deg, e);
    gcn_dinv<<<blk_n, T, 0, stream>>>(deg, dinv, n);

    // layer 1: lin = x@W1 ; agg seeded with self-loop ; edge scatter
    gcn_gemm128<false><<<blk_gemm, T, 0, stream>>>(x, nullptr, W1, bufA, bufB, dinv, n);
    gcn_scatter<<<blk_ew, T, 0, stream>>>(srcp, dstp, dinv, bufA, bufB, e);

    // layer 2: reads relu(bufB + b1) on the fly; in-place agg seed is safe
    // (each wave fully reads its 16 rows before writing them)
    gcn_gemm128<true><<<blk_gemm, T, 0, stream>>>(bufB, b1, W2, bufA, bufB, dinv, n);
    gcn_scatter<<<blk_ew, T, 0, stream>>>(srcp, dstp, dinv, bufA, bufB, e);

    // output projection (fused relu(h + b2)) + softmax
    gcn_out_softmax<<<blk_ow, T, 0, stream>>>(bufB, b2, Wo, bo, outp, n);
}